// SCSA_5995774345639
// MI455X (gfx1250) — compile-verified
//
#include <hip/hip_runtime.h>

typedef __attribute__((ext_vector_type(2))) float v2f;
typedef __attribute__((ext_vector_type(8))) float v8f;

#define B_   32
#define C_   256
#define H_   56
#define W_   56
#define HW_  3136
#define CG_  64
#define NT_  49
#define NP_  64
#define EPSV 1e-5f

// workspace offsets (in floats)
#define OFF_XH 0
#define OFF_XW 458752
#define OFF_AH 917504
#define OFF_AW 1376256
#define OFF_XP 1835008
#define OFF_XG 2359296
#define OFF_Q  2883584
#define OFF_K  3407872
#define OFF_V  3932160
#define OFF_MC 4456448

// ---------------------------------------------------------------- K1: x_h, x_w
__global__ void k_pool(const float* __restrict__ x, float* __restrict__ ws) {
  __shared__ float pl[HW_];
  int bc = blockIdx.x;                       // b*C + c
  const float* xp = x + (size_t)bc * HW_;
  for (int i = threadIdx.x; i < HW_; i += blockDim.x) pl[i] = xp[i];
  __syncthreads();
  int t = threadIdx.x;
  if (t < H_) {                              // mean over W -> x_h
    float s = 0.f;
    for (int w = 0; w < W_; ++w) s += pl[t * W_ + w];
    ws[OFF_XH + bc * H_ + t] = s * (1.0f / W_);
  } else if (t >= 64 && t < 64 + W_) {       // mean over H -> x_w
    int w = t - 64; float s = 0.f;
    for (int h = 0; h < H_; ++h) s += pl[h * W_ + w];
    ws[OFF_XW + bc * W_ + w] = s * (1.0f / H_);
  }
}

// ------------------------------------------- K2: dwconv + GroupNorm(4) + sigmoid
__global__ void k_convgn(const float* __restrict__ wh0, const float* __restrict__ wh1,
                         const float* __restrict__ wh2, const float* __restrict__ wh3,
                         const float* __restrict__ ww0, const float* __restrict__ ww1,
                         const float* __restrict__ ww2, const float* __restrict__ ww3,
                         const float* __restrict__ gnh_g, const float* __restrict__ gnh_b,
                         const float* __restrict__ gnw_g, const float* __restrict__ gnw_b,
                         float* __restrict__ ws) {
  __shared__ float inb[CG_ * 56];
  __shared__ float cvb[CG_ * 56];
  __shared__ float wsm[CG_ * 9];
  __shared__ float red[256], red2[256];
  __shared__ float mv[2];
  int bid = blockIdx.x;
  int g   = bid & 3;
  int dir = (bid >> 2) & 1;
  int b   = bid >> 3;
  const int ksz[4] = {3, 5, 7, 9};
  int k = ksz[g];
  const float* wp = dir ? (g == 0 ? ww0 : g == 1 ? ww1 : g == 2 ? ww2 : ww3)
                        : (g == 0 ? wh0 : g == 1 ? wh1 : g == 2 ? wh2 : wh3);
  const float* gam = dir ? gnw_g : gnh_g;
  const float* bet = dir ? gnw_b : gnh_b;
  const float* src = ws + (dir ? OFF_XW : OFF_XH);
  float* dst       = ws + (dir ? OFF_AW : OFF_AH);
  const int L = 56;
  for (int e = threadIdx.x; e < CG_ * L; e += blockDim.x) {
    int cl = e / L, p = e - cl * L;
    inb[e] = src[(b * C_ + g * CG_ + cl) * L + p];
  }
  for (int e = threadIdx.x; e < CG_ * k; e += blockDim.x) wsm[e] = wp[e];
  __syncthreads();
  float s = 0.f, s2 = 0.f;
  int hk = k >> 1;
  for (int e = threadIdx.x; e < CG_ * L; e += blockDim.x) {
    int cl = e / L, p = e - cl * L;
    float acc = 0.f;
    for (int j = 0; j < k; ++j) {
      int q = p + j - hk;
      if (q >= 0 && q < L) acc += inb[cl * L + q] * wsm[cl * k + j];
    }
    cvb[e] = acc;
    s += acc; s2 += acc * acc;
  }
  red[threadIdx.x] = s; red2[threadIdx.x] = s2;
  __syncthreads();
  for (int off = 128; off > 0; off >>= 1) {
    if ((int)threadIdx.x < off) {
      red[threadIdx.x]  += red[threadIdx.x + off];
      red2[threadIdx.x] += red2[threadIdx.x + off];
    }
    __syncthreads();
  }
  if (threadIdx.x == 0) {
    float inv = 1.0f / (float)(CG_ * L);
    float m = red[0] * inv;
    float v = red2[0] * inv - m * m;
    mv[0] = m; mv[1] = rsqrtf(v + EPSV);
  }
  __syncthreads();
  float m = mv[0], rs = mv[1];
  for (int e = threadIdx.x; e < CG_ * L; e += blockDim.x) {
    int cl = e / L, p = e - cl * L;
    int c = g * CG_ + cl;
    float z = (cvb[e] - m) * rs * gam[c] + bet[c];
    dst[(b * C_ + c) * L + p] = 1.0f / (1.0f + __expf(-z));
  }
}

// --------------------------------------- K3: gated 8x8 window means -> xp[C][64]
__global__ void k_gatepool(const float* __restrict__ x, float* __restrict__ ws) {
  __shared__ float ah[H_], aw[W_], pool[64];
  int bc = blockIdx.x;
  if (threadIdx.x < 64) pool[threadIdx.x] = 0.f;
  if (threadIdx.x >= 64 && threadIdx.x < 64 + H_)
    ah[threadIdx.x - 64] = ws[OFF_AH + bc * H_ + (threadIdx.x - 64)];
  if (threadIdx.x >= 128 && threadIdx.x < 128 + W_)
    aw[threadIdx.x - 128] = ws[OFF_AW + bc * W_ + (threadIdx.x - 128)];
  __syncthreads();
  const float* xp = x + (size_t)bc * HW_;
  for (int i = threadIdx.x; i < HW_; i += blockDim.x) {
    int h = i / W_, w = i - h * W_;
    float gv = xp[i] * ah[h] * aw[w];
    atomicAdd(&pool[(h >> 3) * 7 + (w >> 3)], gv);
  }
  __syncthreads();
  if (threadIdx.x < NP_) {
    float v = (threadIdx.x < NT_) ? pool[threadIdx.x] * (1.0f / 64.0f) : 0.f;
    ws[OFF_XP + bc * NP_ + threadIdx.x] = v;
  }
}

// ------------------------------------------------- K4: GroupNorm(groups=1) -> xg
__global__ void k_gn1(const float* __restrict__ g1, const float* __restrict__ b1,
                      float* __restrict__ ws) {
  __shared__ float red[256], red2[256], mv[2];
  int b = blockIdx.x;
  const float* xp = ws + OFF_XP + b * C_ * NP_;
  float s = 0.f, s2 = 0.f;
  for (int idx = threadIdx.x; idx < C_ * NT_; idx += blockDim.x) {
    int c = idx / NT_, n = idx - c * NT_;
    float v = xp[c * NP_ + n];
    s += v; s2 += v * v;
  }
  red[threadIdx.x] = s; red2[threadIdx.x] = s2;
  __syncthreads();
  for (int off = 128; off > 0; off >>= 1) {
    if ((int)threadIdx.x < off) {
      red[threadIdx.x]  += red[threadIdx.x + off];
      red2[threadIdx.x] += red2[threadIdx.x + off];
    }
    __syncthreads();
  }
  if (threadIdx.x == 0) {
    float inv = 1.0f / (float)(C_ * NT_);
    float m = red[0] * inv;
    float v = red2[0] * inv - m * m;
    mv[0] = m; mv[1] = rsqrtf(v + EPSV);
  }
  __syncthreads();
  float m = mv[0], rs = mv[1];
  float* xg = ws + OFF_XG + b * C_ * NP_;
  for (int idx = threadIdx.x; idx < C_ * NP_; idx += blockDim.x) {
    int c = idx / NP_, n = idx - c * NP_;
    float v = (n < NT_) ? ((xp[c * NP_ + n] - m) * rs * g1[c] + b1[c]) : 0.f;
    xg[idx] = v;
  }
}

// -------------------------------- K5: q/k/v = W @ xg  (WMMA f32 16x16x4, 1 wave)
__global__ void k_qkv(const float* __restrict__ wq, const float* __restrict__ wk,
                      const float* __restrict__ wv, float* __restrict__ ws) {
  int bi  = blockIdx.x;
  int nt  = bi & 3;
  int mt  = (bi >> 2) & 15;
  int b   = (bi >> 6) & 31;
  int mat = bi >> 11;
  const float* Wm = (mat == 0) ? wq : (mat == 1) ? wk : wv;
  const float* X  = ws + OFF_XG + b * C_ * NP_;
  float* Y = ws + ((mat == 0) ? OFF_Q : (mat == 1) ? OFF_K : OFF_V) + b * C_ * NP_;
  int lane = threadIdx.x;
  int hl   = lane >> 4;
  int l16  = lane & 15;
  int m0 = mt * 16, n0 = nt * 16;
  v8f acc = {};
  for (int kk = 0; kk < C_; kk += 4) {
    int ka = kk + 2 * hl;
    v2f a, bf;
    a.x  = Wm[(m0 + l16) * C_ + ka];
    a.y  = Wm[(m0 + l16) * C_ + ka + 1];
    bf.x = X[ka * NP_ + n0 + l16];
    bf.y = X[(ka + 1) * NP_ + n0 + l16];
    acc = __builtin_amdgcn_wmma_f32_16x16x4_f32(false, a, false, bf, (short)0, acc,
                                                false, false);
  }
#pragma unroll
  for (int r = 0; r < 8; ++r)
    Y[(m0 + r + 8 * hl) * NP_ + n0 + l16] = acc[r];
}

// ----------------- K6: s=q·kT/7 (WMMA) -> softmax -> ca=attn·v (WMMA) -> Mc
__global__ void k_attn(float* __restrict__ ws) {
  __shared__ float sL[16 * 256];
  __shared__ float caL[16 * NP_];
  __shared__ float red[256];
  __shared__ float rmax[16], rsum[16];
  int b  = blockIdx.x >> 4;
  int mt = blockIdx.x & 15;
  const float* Q  = ws + OFF_Q + b * C_ * NP_;
  const float* Km = ws + OFF_K + b * C_ * NP_;
  const float* V  = ws + OFF_V + b * C_ * NP_;
  int tid  = threadIdx.x;
  int wave = tid >> 5;
  int lane = tid & 31;
  int hl   = lane >> 4;
  int l16  = lane & 15;
  int m0 = mt * 16;
  // phase 1: scores (rows m0..m0+15, all 256 cols), K dim = 64 padded tokens
  for (int sub = 0; sub < 2; ++sub) {
    int nt = wave * 2 + sub;
    v8f acc = {};
    for (int kk = 0; kk < NP_; kk += 4) {
      int ka = kk + 2 * hl;
      v2f a, bf;
      a.x  = Q[(m0 + l16) * NP_ + ka];
      a.y  = Q[(m0 + l16) * NP_ + ka + 1];
      bf.x = Km[(nt * 16 + l16) * NP_ + ka];
      bf.y = Km[(nt * 16 + l16) * NP_ + ka + 1];
      acc = __builtin_amdgcn_wmma_f32_16x16x4_f32(false, a, false, bf, (short)0, acc,
                                                  false, false);
    }
#pragma unroll
    for (int r = 0; r < 8; ++r)
      sL[(r + 8 * hl) * 256 + nt * 16 + l16] = acc[r] * (1.0f / 7.0f);
  }
  __syncthreads();
  // phase 2: row softmax over 256 (16 threads per row)
  int r = tid >> 4, j0 = tid & 15;
  float pm = -3.0e38f;
  for (int j = j0; j < 256; j += 16) pm = fmaxf(pm, sL[r * 256 + j]);
  red[tid] = pm;
  __syncthreads();
  if (j0 == 0) {
    float mx = red[r * 16];
    for (int j = 1; j < 16; ++j) mx = fmaxf(mx, red[r * 16 + j]);
    rmax[r] = mx;
  }
  __syncthreads();
  float mx = rmax[r];
  float ps = 0.f;
  for (int j = j0; j < 256; j += 16) {
    float e = __expf(sL[r * 256 + j] - mx);
    sL[r * 256 + j] = e;
    ps += e;
  }
  red[tid] = ps;
  __syncthreads();
  if (j0 == 0) {
    float sm = 0.f;
    for (int j = 0; j < 16; ++j) sm += red[r * 16 + j];
    rsum[r] = sm;
  }
  __syncthreads();
  float inv = 1.0f / rsum[r];
  for (int j = j0; j < 256; j += 16) sL[r * 256 + j] *= inv;
  __syncthreads();
  // phase 3: ca = attn @ v  (waves 0..3 each own one 16-col tile; K = 256)
  if (wave < 4) {
    int nt = wave;
    v8f acc = {};
    for (int kk = 0; kk < C_; kk += 4) {
      int ka = kk + 2 * hl;
      v2f a, bf;
      a.x  = sL[l16 * 256 + ka];
      a.y  = sL[l16 * 256 + ka + 1];
      bf.x = V[ka * NP_ + nt * 16 + l16];
      bf.y = V[(ka + 1) * NP_ + nt * 16 + l16];
      acc = __builtin_amdgcn_wmma_f32_16x16x4_f32(false, a, false, bf, (short)0, acc,
                                                  false, false);
    }
#pragma unroll
    for (int rr = 0; rr < 8; ++rr)
      caL[(rr + 8 * hl) * NP_ + nt * 16 + l16] = acc[rr];
  }
  __syncthreads();
  // phase 4: Mc = sigmoid(mean over 49 tokens)
  if (tid < 16) {
    float s = 0.f;
    for (int n = 0; n < NT_; ++n) s += caL[tid * NP_ + n];
    ws[OFF_MC + b * C_ + m0 + tid] = 1.0f / (1.0f + __expf(-s * (1.0f / NT_)));
  }
}

// ---------------------------------------------- K7: out = x * Ah * Aw * Mc
__global__ void k_out(const float* __restrict__ x, const float* __restrict__ ws,
                      float* __restrict__ out) {
  __shared__ float ah[H_], aw[W_];
  __shared__ float mcs;
  int bc = blockIdx.x;
  if (threadIdx.x < H_) ah[threadIdx.x] = ws[OFF_AH + bc * H_ + threadIdx.x];
  if (threadIdx.x >= 64 && threadIdx.x < 64 + W_) {
    int w = threadIdx.x - 64;
    aw[w] = ws[OFF_AW + bc * W_ + w];
  }
  if (threadIdx.x == 255) mcs = ws[OFF_MC + bc];
  __syncthreads();
  const float* xp = x + (size_t)bc * HW_;
  float* op = out + (size_t)bc * HW_;
  float mc = mcs;
  for (int i = threadIdx.x; i < HW_; i += blockDim.x) {
    int h = i / W_, w = i - h * W_;
    op[i] = xp[i] * ah[h] * aw[w] * mc;
  }
}

extern "C" void kernel_launch(void* const* d_in, const int* in_sizes, int n_in,
                              void* d_out, int out_size, void* d_ws, size_t ws_size,
                              hipStream_t stream) {
  const float* x     = (const float*)d_in[0];
  const float* wh0   = (const float*)d_in[1];
  const float* wh1   = (const float*)d_in[2];
  const float* wh2   = (const float*)d_in[3];
  const float* wh3   = (const float*)d_in[4];
  const float* ww0   = (const float*)d_in[5];
  const float* ww1   = (const float*)d_in[6];
  const float* ww2   = (const float*)d_in[7];
  const float* ww3   = (const float*)d_in[8];
  const float* gnh_g = (const float*)d_in[9];
  const float* gnh_b = (const float*)d_in[10];
  const float* gnw_g = (const float*)d_in[11];
  const float* gnw_b = (const float*)d_in[12];
  const float* gn1_g = (const float*)d_in[13];
  const float* gn1_b = (const float*)d_in[14];
  const float* wq    = (const float*)d_in[15];
  const float* wk    = (const float*)d_in[16];
  const float* wv    = (const float*)d_in[17];
  float* ws  = (float*)d_ws;
  float* out = (float*)d_out;

  k_pool<<<B_ * C_, 256, 0, stream>>>(x, ws);
  k_convgn<<<B_ * 8, 256, 0, stream>>>(wh0, wh1, wh2, wh3, ww0, ww1, ww2, ww3,
                                       gnh_g, gnh_b, gnw_g, gnw_b, ws);
  k_gatepool<<<B_ * C_, 256, 0, stream>>>(x, ws);
  k_gn1<<<B_, 256, 0, stream>>>(gn1_g, gn1_b, ws);
  k_qkv<<<3 * B_ * 16 * 4, 32, 0, stream>>>(wq, wk, wv, ws);
  k_attn<<<B_ * 16, 256, 0, stream>>>(ws);
  k_out<<<B_ * C_, 256, 0, stream>>>(x, ws, out);
}